// MMD_53008486367839
// MI455X (gfx1250) — compile-verified
//
#include <hip/hip_runtime.h>
#include <hip/hip_bf16.h>

typedef float v2f __attribute__((ext_vector_type(2)));
typedef float v8f __attribute__((ext_vector_type(8)));

#define BATCH 64
#define NPTS  1024
#define TOT   2048          // X and Y unified into one signed-weight point set
#define TILES (TOT / 16)    // 128 tiles per dimension

// Load point i of batch b from the unified set:
// i <  NPTS -> X point, sign = +lifetime
// i >= NPTS -> Y point, sign = -lifetime
// Returns (birth, lifetime, birth^2+lifetime^2, signed_weight)
__device__ __forceinline__ float4 load_point(const float2* __restrict__ Xp,
                                             const float2* __restrict__ Yp,
                                             int b, int i) {
    const float2* src = (i < NPTS) ? (Xp + (size_t)b * NPTS + i)
                                   : (Yp + (size_t)b * NPTS + (i - NPTS));
    float2 p = *src;                    // (birth, death) -> global_load_b64
    float life = p.y - p.x;
    float sq   = fmaf(p.x, p.x, life * life);
    float s    = (i < NPTS) ? life : -life;
    return make_float4(p.x, life, sq, s);
}

__global__ void __launch_bounds__(256)
mmd_wmma_kernel(const float2* __restrict__ X, const float2* __restrict__ Y,
                const float*  __restrict__ W, const float*  __restrict__ NS,
                float* __restrict__ out) {
    const int lane = threadIdx.x & 31;
    const int wave = threadIdx.x >> 5;                  // 0..7
    const int b    = blockIdx.x >> 4;                   // 64 batches
    // wave-uniform: pin to SGPR so loop control is scalar
    const int rowblk = __builtin_amdgcn_readfirstlane(
        ((blockIdx.x & 15) << 3) + wave);               // 0..127 (16-row block)
    const int half = lane >> 4;                         // K/row half split
    const int lid  = lane & 15;

    // ---- stage the batch's points into LDS, pre-fragmented for the B side ----
    // bvec[i] = (-2p0, -2p1, 1, |p|^2): lanes pick .xy (K=0,1) or .zw (K=2,3)
    // directly as the WMMA B fragment. sw[i] = signed weight.
    __shared__ float4 bvec[TOT];   // 32 KB
    __shared__ float  sw[TOT];     //  8 KB
    for (int i = threadIdx.x; i < TOT; i += 256) {
        float4 p = load_point(X, Y, b, i);
        bvec[i] = make_float4(-2.0f * p.x, -2.0f * p.y, 1.0f, p.z);
        sw[i]   = p.w;
    }
    __syncthreads();

    // ---- A fragment: augmented row vector (x0, x1, |x|^2, 1) ----
    // lanes 0-15: K=0,1 -> (x0, x1); lanes 16-31: K=2,3 -> (|x|^2, 1)
    const int row = (rowblk << 4) + lid;
    float4 pa = load_point(X, Y, b, row);
    v2f afrag;
    afrag.x = half ? pa.z : pa.x;
    afrag.y = half ? 1.0f : pa.y;
    const float s_self = pa.w;   // signed weight of row (lane%16) of this block

    float acc[8];
    #pragma unroll
    for (int r = 0; r < 8; ++r) acc[r] = 0.0f;

    const float NL2E = -1.4426950408889634f;  // -log2(e); WIDTH == 1

    // One 16x16 tile: B fragment straight from LDS, d^2 from one WMMA,
    // exp + weighted accumulate. wscale folds the symmetry factor.
    // No max(d2,0) clamp: d2 >= 0 up to ~1ulp cancellation noise, and
    // exp(+eps) deviates from the clamped value by ~1e-6 relative, only for
    // near-duplicate pairs — far below tolerance. (The clamp was costing
    // 16 VALU/tile as canonicalize+max.)
    auto do_tile = [&](int j, float wscale) {
        const int col = (j << 4) + lid;
        v2f bfrag = *(const v2f*)((const float*)&bvec[col] + (half << 1)); // ds_load_b64
        const float s_col = sw[col] * wscale;                              // ds_load_b32

        v8f c = {0.f, 0.f, 0.f, 0.f, 0.f, 0.f, 0.f, 0.f};
        v8f d2 = __builtin_amdgcn_wmma_f32_16x16x4_f32(
            false, afrag, false, bfrag, (short)0, c, false, false);

        #pragma unroll
        for (int r = 0; r < 8; ++r) {
            float k = __builtin_amdgcn_exp2f(NL2E * d2[r]);  // exp(-d2)
            acc[r] = fmaf(k, s_col, acc[r]);                 // fold column weight
        }
    };

    // ---- symmetric sweep: diagonal tile once at half weight, j>rowblk twice
    // (restored by the global x2 at the end). ----
    do_tile(rowblk, 0.5f);
    #pragma unroll 2
    for (int j = rowblk + 1; j < TILES; ++j)
        do_tile(j, 1.0f);

    // ---- epilogue: apply row weights via lane shuffle, then wave reduce ----
    // D VGPR r corresponds to row (8*half + r); that row's weight lives in
    // lane (8*half + r) as s_self.
    float v = 0.0f;
    #pragma unroll
    for (int r = 0; r < 8; ++r) {
        float srow = __shfl(s_self, (half << 3) + r, 32);
        v = fmaf(acc[r], srow, v);
    }
    #pragma unroll
    for (int off = 16; off; off >>= 1)
        v += __shfl_xor(v, off, 32);

    if (lane == 0) {
        float ns = NS[b];
        // x2 restores the symmetric double-count (diagonal was pre-halved)
        atomicAdd(out, 2.0f * v * W[b] / (ns * ns));
    }
}

extern "C" void kernel_launch(void* const* d_in, const int* in_sizes, int n_in,
                              void* d_out, int out_size, void* d_ws, size_t ws_size,
                              hipStream_t stream) {
    const float2* X  = (const float2*)d_in[0];  // [B,1024,2] f32
    const float2* Y  = (const float2*)d_in[1];  // [B,1024,2] f32
    const float*  W  = (const float*)d_in[2];   // [B]
    const float*  NS = (const float*)d_in[3];   // [B]
    float* out = (float*)d_out;                 // scalar f32

    hipMemsetAsync(out, 0, sizeof(float) * (size_t)out_size, stream);

    dim3 grid(BATCH * 16);   // 64 batches x 16 blocks; 8 waves/block -> 128 row blocks
    dim3 block(256);
    mmd_wmma_kernel<<<grid, block, 0, stream>>>(X, Y, W, NS, out);
}